// ROIPooling_19292993093737
// MI455X (gfx1250) — compile-verified
//
#include <hip/hip_runtime.h>
#include <stdint.h>

// ROI max-pooling for MI455X (gfx1250).
// Grid: one 64-thread block (2 x wave32) per ROI. Each thread owns a float4
// channel slice (64 threads x 4 = 256 channels), so each (row,col) position of
// the ROI window is one fully coalesced 1KB transaction.
// Data path: CDNA5 async loads (global_load_async_to_lds_b128, ASYNCcnt) into a
// 4-slot LDS ring, constant-depth-3 software pipeline with s_wait_asynccnt.

#define PH 7
#define PW 7
#define FH 64
#define FW 64
#define FC 256
#define THREADS 64      // FC / 4 channels per thread
#define SLOT_BYTES 1024 // THREADS * 16 bytes

typedef float vfloat4 __attribute__((ext_vector_type(4)));

struct AxisParams { int start, step, length; };

// Reproduces the reference's _axis_bins start/step/length derivation.
// All float ops use non-contracted intrinsics so the int truncation matches
// the two-step float math of the reference exactly.
__device__ __forceinline__ AxisParams axis_params(float center, float size,
                                                  int limit, int pooled) {
    float half = __fmul_rn(size, 0.5f);
    int start = (int)__fsub_rn(center, half);          // trunc toward zero
    if (start < 0) start = 0;
    int end = (int)__fadd_rn(center, half);
    if (end > limit) end = limit;
    int pad_raw = pooled - (end - start);
    if (pad_raw > 0) {
        int pad = (pad_raw + 1) >> 1;                  // ceil(pad_raw/2), pad_raw>0
        int top = limit - end;
        int bottom = start;
        int ns, ne;
        if (top < pad)          { ns = start - (2 * pad - top);  ne = end + top; }
        else if (bottom < pad)  { ns = start - bottom;           ne = end + (2 * pad - bottom); }
        else                    { ns = start - pad;              ne = end + pad; }
        start = ns; end = ne;
    }
    AxisParams a;
    a.start  = start;
    a.length = end - start;                            // >= pooled by construction
    a.step   = a.length / pooled;
    if (a.step < 1) a.step = 1;
    return a;
}

__device__ __forceinline__ int iclamp(int v, int lo, int hi) {
    return v < lo ? lo : (v > hi ? hi : v);
}

__global__ __launch_bounds__(THREADS)
void roi_maxpool_kernel(const float* __restrict__ fm,
                        const float* __restrict__ rois,
                        float* __restrict__ out,
                        int rois_per_batch) {
    __shared__ vfloat4 stage[4][THREADS];              // 4 KB ring, 4 slots x 1KB

    const int roi = blockIdx.x;
    const int b   = roi / rois_per_batch;
    const int tid = threadIdx.x;

    const vfloat4 rv = reinterpret_cast<const vfloat4*>(rois)[roi]; // (x,y,w,h)
    const AxisParams ay = axis_params(rv.y, rv.w, FH, PH);          // rows
    const AxisParams ax = axis_params(rv.x, rv.z, FW, PW);          // cols

    const float* fbase = fm + ((size_t)b * FH * FW * FC) + tid * 4;
    float*       obase = out + ((size_t)roi * PH * PW * FC) + tid * 4;

    // 32-bit LDS byte address of this lane's slot-0 element (low 32 bits of the
    // flat shared pointer are the LDS address per the CDNA5 aperture mapping).
    const unsigned lds0 = (unsigned)(uintptr_t)&stage[0][tid];

    const float NEG_INF = -__builtin_inff();

    for (int ph = 0; ph < PH; ++ph) {
        const int r0   = iclamp(ay.start + ph * ay.step, 0, FH);
        const int rend = (ph == PH - 1) ? ay.length : (ph + 1) * ay.step;
        const int r1   = iclamp(ay.start + rend, 0, FH);
        for (int pw = 0; pw < PW; ++pw) {
            const int c0   = iclamp(ax.start + pw * ax.step, 0, FW);
            const int cend = (pw == PW - 1) ? ax.length : (pw + 1) * ax.step;
            const int c1   = iclamp(ax.start + cend, 0, FW);

            vfloat4 acc = {NEG_INF, NEG_INF, NEG_INF, NEG_INF};

            const int nr = r1 - r0;
            const int nc = c1 - c0;
            const int n  = (nr > 0 && nc > 0) ? nr * nc : 0;

            if (n > 0) {
                int ri = r0, ci = c0;                  // saturating issue cursor
                // Prologue: prime 3 positions (cursor saturates on tiny bins;
                // redundant re-loads of the last position are harmless).
                #pragma unroll
                for (int j = 0; j < 3; ++j) {
                    const float* g = fbase + (size_t)(ri * FW + ci) * FC;
                    unsigned loff = lds0 + (unsigned)(j * SLOT_BYTES);
                    asm volatile("global_load_async_to_lds_b128 %0, %1, off"
                                 :: "v"(loff), "v"(g) : "memory");
                    if (ci + 1 < c1)      ++ci;
                    else if (ri + 1 < r1) { ++ri; ci = c0; }
                }
                for (int i = 0; i < n; ++i) {
                    // Issue position i+3 into slot (i+3)&3 (dummy when past end).
                    const float* g = fbase + (size_t)(ri * FW + ci) * FC;
                    unsigned loff = lds0 + (unsigned)((((unsigned)i + 3u) & 3u) * SLOT_BYTES);
                    asm volatile("global_load_async_to_lds_b128 %0, %1, off"
                                 :: "v"(loff), "v"(g) : "memory");
                    if (ci + 1 < c1)      ++ci;
                    else if (ri + 1 < r1) { ++ri; ci = c0; }

                    // Async loads complete in order: <=3 outstanding means
                    // position i has landed in its LDS slot.
                    asm volatile("s_wait_asynccnt 0x3" ::: "memory");

                    vfloat4 v = stage[i & 3][tid];
                    acc.x = fmaxf(acc.x, v.x);
                    acc.y = fmaxf(acc.y, v.y);
                    acc.z = fmaxf(acc.z, v.z);
                    acc.w = fmaxf(acc.w, v.w);
                }
                // Drain trailing dummy loads so the next bin's slot writes
                // cannot race with this bin's outstanding async writes.
                asm volatile("s_wait_asynccnt 0x0" ::: "memory");
            }

            // Output is written once and never re-read: bypass L2 residency
            // (NT hint) so the L2-resident feature map stays hot.
            __builtin_nontemporal_store(acc,
                reinterpret_cast<vfloat4*>(obase + (ph * PW + pw) * FC));
        }
    }
}

extern "C" void kernel_launch(void* const* d_in, const int* in_sizes, int n_in,
                              void* d_out, int out_size, void* d_ws, size_t ws_size,
                              hipStream_t stream) {
    (void)n_in; (void)out_size; (void)d_ws; (void)ws_size;
    const float* fm   = (const float*)d_in[0];
    const float* rois = (const float*)d_in[1];
    float*       out  = (float*)d_out;

    const int total_rois = in_sizes[1] / 4;               // B*N
    int B = in_sizes[0] / (FH * FW * FC);
    if (B < 1) B = 1;
    const int rois_per_batch = total_rois / B;

    roi_maxpool_kernel<<<dim3(total_rois), dim3(THREADS), 0, stream>>>(
        fm, rois, out, rois_per_batch);
}